// OpticalFlowConsistencyLoss_45878840656470
// MI455X (gfx1250) — compile-verified
//
#include <hip/hip_runtime.h>
#include <cstdint>
#include <cstddef>

// ---------------- problem constants ----------------
#define BATCH   16
#define HH      512
#define WW0     512
#define SH      128      // WH
#define SW      128      // WW
#define NOFF    13       // 2*RADIUS+1
#define NOFF2   169      // NOFF*NOFF
#define PR      10       // (PATCH-1)/2 band half-width
#define N128    (BATCH*SH*SW)      // 262144
#define N512    (BATCH*HH*WW0)     // 4194304

typedef float v2f __attribute__((ext_vector_type(2)));
typedef float v8f __attribute__((ext_vector_type(8)));

__device__ __forceinline__ float logw(float v) { return log1pf(fmaxf(v, 0.0f)); }
__device__ __forceinline__ int   iclamp(int v, int lo, int hi) { return v < lo ? lo : (v > hi ? hi : v); }

__device__ __forceinline__ v8f wmma_f32(v2f a, v2f b, v8f c) {
  return __builtin_amdgcn_wmma_f32_16x16x4_f32(false, a, false, b, (short)0, c, false, false);
}

// ---------------- stage 1: log1p + exact 4x downsample + rain ----------------
// 512->128 bilinear taps collapse to 0.25*(rows 4i+1,4i+2 x cols 4j+1,4j+2)
__global__ void prep_kernel(const float* __restrict__ ref, const float* __restrict__ target,
                            float* __restrict__ prevw, float* __restrict__ currw,
                            float* __restrict__ rain) {
  int g = blockIdx.x * blockDim.x + threadIdx.x;
  if (g >= N128) return;
  int b = g / (SH * SW), rem = g % (SH * SW);
  int i = rem / SW, j = rem % SW;
  const float* rb = ref    + (size_t)b * HH * WW0;
  const float* tb = target + (size_t)b * HH * WW0;
  int i0 = 4 * i + 1, i1 = 4 * i + 2, j0 = 4 * j + 1, j1 = 4 * j + 2;
  prevw[g] = 0.25f * (logw(rb[i0*WW0+j0]) + logw(rb[i0*WW0+j1]) +
                      logw(rb[i1*WW0+j0]) + logw(rb[i1*WW0+j1]));
  currw[g] = 0.25f * (logw(tb[i0*WW0+j0]) + logw(tb[i0*WW0+j1]) +
                      logw(tb[i1*WW0+j0]) + logw(tb[i1*WW0+j1]));
  float tr = 0.25f * (tb[i0*WW0+j0] + tb[i0*WW0+j1] + tb[i1*WW0+j0] + tb[i1*WW0+j1]);
  rain[g] = (tr >= 0.1f) ? 1.0f : 0.0f;
}

// ---------------- separable 7-tap gaussian (sigma=1, clamped edges) ----------------
__device__ __forceinline__ void gauss7(float* w) {
  float s = 0.0f;
#pragma unroll
  for (int t = 0; t < 7; ++t) { float c = (float)(t - 3); w[t] = expf(-(c * c) * 0.5f); s += w[t]; }
#pragma unroll
  for (int t = 0; t < 7; ++t) w[t] /= s;
}

__global__ void blur_h_kernel(const float* __restrict__ src, float* __restrict__ dst) {
  int g = blockIdx.x * blockDim.x + threadIdx.x;
  if (g >= N128) return;
  float w[7]; gauss7(w);
  int b = g / (SH * SW), rem = g % (SH * SW);
  int i = rem / SW, j = rem % SW;
  const float* row = src + (size_t)b * SH * SW + (size_t)i * SW;
  float acc = 0.0f;
#pragma unroll
  for (int t = 0; t < 7; ++t) acc += w[t] * row[iclamp(j + t - 3, 0, SW - 1)];
  dst[g] = acc;
}

__global__ void blur_v_kernel(const float* __restrict__ src, float* __restrict__ dst,
                              const float* __restrict__ mask) {
  int g = blockIdx.x * blockDim.x + threadIdx.x;
  if (g >= N128) return;
  float w[7]; gauss7(w);
  int b = g / (SH * SW), rem = g % (SH * SW);
  int i = rem / SW, j = rem % SW;
  const float* img = src + (size_t)b * SH * SW;
  float acc = 0.0f;
#pragma unroll
  for (int t = 0; t < 7; ++t) acc += w[t] * img[(size_t)iclamp(i + t - 3, 0, SH - 1) * SW + j];
  if (mask) acc *= mask[g];
  dst[g] = acc;
}

// ---------------- stage 2: init argmin keys ----------------
__global__ void init_best_kernel(unsigned long long* __restrict__ best) {
  int g = blockIdx.x * blockDim.x + threadIdx.x;
  if (g < N128) best[g] = 0xFFFFFFFFFFFFFFFFull;
}

// ---------------- stage 3: WMMA cost kernel ----------------
// P = S * D * S  (S = banded 0/1, |i-j|<=10) == 21x21 zero-padded box sum of D.
// One block per (offset o, batch b). 8 waves; wave w owns output row-tile w.
// Band fragments of S are precomputed per wave: they depend only on
// (kt - tile) in {-1,0,+1} and the k-step, and serve both GEMMs by symmetry.
__global__ __launch_bounds__(256) void cost_kernel(const float* __restrict__ prevw,
                                                   const float* __restrict__ currw,
                                                   unsigned long long* __restrict__ best) {
  extern __shared__ float smem[];
  constexpr int ST = 130;                 // padded LDS row stride
  float* Dm = smem;                       // 128 x 130
  float* Tm = smem + SH * ST;             // 128 x 130
  const int o = blockIdx.x, b = blockIdx.y;
  const int dy = o / NOFF - 6, dx = o % NOFF - 6;
  const float* P = prevw + (size_t)b * SH * SW;
  const float* C = currw + (size_t)b * SH * SW;
  const int tid = threadIdx.x;

  // Build D = (prev - shifted_curr)^2 in LDS
  for (int e = 0; e < 64; ++e) {
    int lin = e * 256 + tid;
    int i = lin >> 7, j = lin & 127;
    int si = iclamp(i + dy, 0, SH - 1), sj = iclamp(j + dx, 0, SW - 1);
    float d = P[lin] - C[si * SW + sj];
    Dm[i * ST + j] = d * d;
  }
  __syncthreads();

  const int lane = tid & 31, w = tid >> 5;
  const int half = lane >> 4, l15 = lane & 15;

  // Precompute the 12 S-band fragments (dt = kt-tile+1 in 0..2, step s in 0..3).
  // Element value = 1 if |k - n| <= PR, with k - n = (dt-1)*16 + 4s + 2*half - l15.
  v2f sband[3][4];
#pragma unroll
  for (int dt = 0; dt < 3; ++dt) {
#pragma unroll
    for (int s = 0; s < 4; ++s) {
      int rel = (dt - 1) * 16 + s * 4 + half * 2 - l15;
      sband[dt][s].x = (rel     >= -PR && rel     <= PR) ? 1.0f : 0.0f;
      sband[dt][s].y = (rel + 1 >= -PR && rel + 1 <= PR) ? 1.0f : 0.0f;
    }
  }

  // GEMM1: T = D * S   (wave w -> rows 16w..16w+15 of T)
  for (int c = 0; c < 8; ++c) {
    v8f acc = {0, 0, 0, 0, 0, 0, 0, 0};
    int kt0 = c > 0 ? c - 1 : 0, kt1 = c < 7 ? c + 1 : 7;   // band: only nearby k-tiles
    for (int kt = kt0; kt <= kt1; ++kt) {
      int dt = kt - c + 1;
#pragma unroll
      for (int s = 0; s < 4; ++s) {
        int ka = kt * 16 + s * 4 + half * 2;
        v2f a;
        a.x = Dm[(w * 16 + l15) * ST + ka];
        a.y = Dm[(w * 16 + l15) * ST + ka + 1];
        acc = wmma_f32(a, sband[dt][s], acc);
      }
    }
#pragma unroll
    for (int v = 0; v < 8; ++v) {
      int row = w * 16 + v + half * 8;
      Tm[row * ST + c * 16 + l15] = acc[v];
    }
  }
  __syncthreads();

  // GEMM2: Pout = S * T, then pooled cost + packed-key atomic argmin
  unsigned long long* bb_ = best + (size_t)b * SH * SW;
  const int kt0 = w > 0 ? w - 1 : 0, kt1 = w < 7 ? w + 1 : 7;
  for (int c = 0; c < 8; ++c) {
    v8f acc = {0, 0, 0, 0, 0, 0, 0, 0};
    for (int kt = kt0; kt <= kt1; ++kt) {
      int dt = kt - w + 1;
#pragma unroll
      for (int s = 0; s < 4; ++s) {
        int ka = kt * 16 + s * 4 + half * 2;
        v2f tb;
        tb.x = Tm[ka * ST + c * 16 + l15];
        tb.y = Tm[(ka + 1) * ST + c * 16 + l15];
        // S symmetric: A[m][k] = band(m,k) uses the same fragment as GEMM1's B
        acc = wmma_f32(sband[dt][s], tb, acc);
      }
    }
#pragma unroll
    for (int v = 0; v < 8; ++v) {
      int row = w * 16 + v + half * 8, col = c * 16 + l15;
      // Positive monotone scale: cannot change the argmin ordering.
      float cost = acc[v] * (1.0f / 441.0f);
      unsigned long long key =
          ((unsigned long long)__float_as_uint(cost) << 32) | (unsigned int)o;
      atomicMin(bb_ + (size_t)row * SW + col, key);   // deterministic argmin, ties->smallest o
    }
  }
}

// ---------------- stage 4: extract raw flow ----------------
__global__ void flow_kernel(const unsigned long long* __restrict__ best,
                            const float* __restrict__ rain,
                            float* __restrict__ fx, float* __restrict__ fy) {
  int g = blockIdx.x * blockDim.x + threadIdx.x;
  if (g >= N128) return;
  unsigned int o = (unsigned int)(best[g] & 0xFFFFFFFFu);
  int dy = (int)(o / NOFF) - 6, dx = (int)(o % NOFF) - 6;
  float r = rain[g];
  fx[g] = (float)dx * r;
  fy[g] = (float)dy * r;
}

// ---------------- stage 5: fused upsample + grid_sample + huber ----------------
__device__ __forceinline__ float up128(const float* __restrict__ f, int y, int x) {
  float sy = fmaxf((y + 0.5f) * 0.25f - 0.5f, 0.0f);
  float sx = fmaxf((x + 0.5f) * 0.25f - 0.5f, 0.0f);
  int y0 = min((int)sy, SH - 1), x0 = min((int)sx, SW - 1);
  int y1 = min(y0 + 1, SH - 1), x1 = min(x0 + 1, SW - 1);
  float wy = sy - (float)y0, wx = sx - (float)x0;
  float v00 = f[y0 * SW + x0], v01 = f[y0 * SW + x1];
  float v10 = f[y1 * SW + x0], v11 = f[y1 * SW + x1];
  return v00 * (1 - wy) * (1 - wx) + v01 * (1 - wy) * wx + v10 * wy * (1 - wx) + v11 * wy * wx;
}

__global__ __launch_bounds__(256) void loss_kernel(const float* __restrict__ pred,
                                                   const float* __restrict__ target,
                                                   const float* __restrict__ ref,
                                                   const float* __restrict__ fxb,
                                                   const float* __restrict__ fyb,
                                                   float* __restrict__ partials) {
  __shared__ float red[256];
  int g = blockIdx.x * 256 + threadIdx.x;   // grid sized exactly to N512
  int b = g / (HH * WW0), rem = g % (HH * WW0);
  int y = rem / WW0, x = rem % WW0;

  const float* fxp = fxb + (size_t)b * SH * SW;
  const float* fyp = fyb + (size_t)b * SH * SW;
  float fx = up128(fxp, y, x) * 4.0f;       // * (X_DIM / WW)
  float fy = up128(fyp, y, x) * 4.0f;

  // grid = base - flow*(2/(D-1));  ix = clip((gx+1)*0.5*(W-1)) == clip(x - fx)
  float ix = fminf(fmaxf((float)x - fx, 0.0f), (float)(WW0 - 1));
  float iy = fminf(fmaxf((float)y - fy, 0.0f), (float)(HH - 1));
  int x0 = min((int)ix, WW0 - 1), y0 = min((int)iy, HH - 1);
  int x1 = min(x0 + 1, WW0 - 1), y1 = min(y0 + 1, HH - 1);
  float wx = ix - (float)x0, wy = iy - (float)y0;
  const float* rb = ref + (size_t)b * HH * WW0;
  float samp = rb[y0 * WW0 + x0] * (1 - wy) * (1 - wx) + rb[y0 * WW0 + x1] * (1 - wy) * wx +
               rb[y1 * WW0 + x0] * wy * (1 - wx)       + rb[y1 * WW0 + x1] * wy * wx;
  float teacher = fmaxf(samp, 0.0f);

  float d = pred[g] - teacher;
  float ad = fabsf(d);
  float loss = (ad < 0.2f) ? 0.5f * d * d : 0.2f * (ad - 0.1f);
  float mask = (teacher > 0.05f || target[g] > 0.05f) ? 1.0f : 0.0f;

  red[threadIdx.x] = loss * mask;
  __syncthreads();
  for (int st = 128; st > 0; st >>= 1) {
    if (threadIdx.x < st) red[threadIdx.x] += red[threadIdx.x + st];
    __syncthreads();
  }
  if (threadIdx.x == 0) partials[blockIdx.x] = red[0];
}

__global__ void reduce_kernel(const float* __restrict__ partials, int n, float* __restrict__ out) {
  __shared__ float red[256];
  float s = 0.0f;
  for (int i = threadIdx.x; i < n; i += 256) s += partials[i];  // fixed order -> deterministic
  red[threadIdx.x] = s;
  __syncthreads();
  for (int st = 128; st > 0; st >>= 1) {
    if (threadIdx.x < st) red[threadIdx.x] += red[threadIdx.x + st];
    __syncthreads();
  }
  if (threadIdx.x == 0) out[0] = red[0] / (float)N512;
}

// ---------------- host launcher ----------------
extern "C" void kernel_launch(void* const* d_in, const int* in_sizes, int n_in,
                              void* d_out, int out_size, void* d_ws, size_t ws_size,
                              hipStream_t stream) {
  const float* pred   = (const float*)d_in[0];
  const float* target = (const float*)d_in[1];
  const float* ref    = (const float*)d_in[2];
  float* out = (float*)d_out;

  float* ws = (float*)d_ws;
  const size_t N = N128;
  float* prevw_raw = ws + 0 * N;
  float* currw_raw = ws + 1 * N;
  float* prevw     = ws + 2 * N;
  float* currw     = ws + 3 * N;
  float* rain      = ws + 4 * N;
  float* tmp       = ws + 5 * N;
  float* fxr       = ws + 6 * N;
  float* fyr       = ws + 7 * N;
  float* fxb       = ws + 8 * N;
  float* fyb       = ws + 9 * N;
  float* partials  = ws + 10 * N;                     // 16384 floats
  unsigned long long* best = (unsigned long long*)(ws + 10 * N + 16384);  // 8B-aligned

  const int blk = 256;
  const int g128 = N128 / blk;       // 1024
  const int g512 = N512 / blk;       // 16384

  prep_kernel<<<g128, blk, 0, stream>>>(ref, target, prevw_raw, currw_raw, rain);

  blur_h_kernel<<<g128, blk, 0, stream>>>(prevw_raw, tmp);
  blur_v_kernel<<<g128, blk, 0, stream>>>(tmp, prevw, nullptr);
  blur_h_kernel<<<g128, blk, 0, stream>>>(currw_raw, tmp);
  blur_v_kernel<<<g128, blk, 0, stream>>>(tmp, currw, nullptr);

  init_best_kernel<<<g128, blk, 0, stream>>>(best);

  size_t lds_bytes = (size_t)2 * SH * 130 * sizeof(float);   // 133120 B (WGP has 320KB)
  cost_kernel<<<dim3(NOFF2, BATCH), blk, lds_bytes, stream>>>(prevw, currw, best);

  flow_kernel<<<g128, blk, 0, stream>>>(best, rain, fxr, fyr);

  blur_h_kernel<<<g128, blk, 0, stream>>>(fxr, tmp);
  blur_v_kernel<<<g128, blk, 0, stream>>>(tmp, fxb, rain);
  blur_h_kernel<<<g128, blk, 0, stream>>>(fyr, tmp);
  blur_v_kernel<<<g128, blk, 0, stream>>>(tmp, fyb, rain);

  loss_kernel<<<g512, blk, 0, stream>>>(pred, target, ref, fxb, fyb, partials);
  reduce_kernel<<<1, blk, 0, stream>>>(partials, g512, out);
}